// RecyclingEncoder_62199716381658
// MI455X (gfx1250) — compile-verified
//
#include <hip/hip_runtime.h>

#define BATCH 8
#define LSEQ 1024
#define HIDDEN 256
#define NUM_RBF 16
#define ROWS 16      // rows per block, phase 1
#define JLANES 16    // j-lanes per row, phase 1

typedef __attribute__((ext_vector_type(2))) float v2f;
typedef __attribute__((ext_vector_type(8))) float v8f;

// -0.5 * log2(e): exp(-0.5*x) == exp2(x * K)
#define NEG_HALF_LOG2E (-0.72134752044448170368f)

// ---------------------------------------------------------------------------
// Phase A: rbf_mean[b,i,r] = (1/L) * sum_j exp(-0.5*(||p_i - p_j|| - c_r)^2)
// One block = one batch b + one 16-row tile. All 1024 positions of the batch
// staged in LDS (12 KB); 16x16 thread layout; 16 f32 accumulators/thread.
// exp folded to a single v_exp_f32 (exp2) with pre-folded constant.
// ---------------------------------------------------------------------------
__global__ __launch_bounds__(256) void rbf_mean_kernel(
    const float* __restrict__ pos,      // [B][L][3]
    const float* __restrict__ centers,  // [NUM_RBF]
    float* __restrict__ rbf_mean)       // [B*L][NUM_RBF]
{
    __shared__ float sPos[LSEQ * 3];                 // 12 KB
    __shared__ float sPart[256][NUM_RBF + 1];        // ~17 KB, padded

    const int tid          = threadIdx.x;
    const int tilesPerB    = LSEQ / ROWS;            // 64
    const int b            = blockIdx.x / tilesPerB;
    const int rt           = blockIdx.x % tilesPerB;

    // stage this batch's positions into LDS (gcd(3,64)=1 -> conflict-free reads)
    const float* posB = pos + (size_t)b * LSEQ * 3;
    for (int idx = tid; idx < LSEQ * 3; idx += 256)
        sPos[idx] = posB[idx];

    float c[NUM_RBF];
#pragma unroll
    for (int r = 0; r < NUM_RBF; ++r) c[r] = centers[r];

    __syncthreads();

    const int jl  = tid & (JLANES - 1);
    const int row = tid >> 4;                        // 0..15
    const int i   = rt * ROWS + row;

    const float xi = sPos[i * 3 + 0];
    const float yi = sPos[i * 3 + 1];
    const float zi = sPos[i * 3 + 2];

    float acc[NUM_RBF];
#pragma unroll
    for (int r = 0; r < NUM_RBF; ++r) acc[r] = 0.0f;

#pragma unroll 2
    for (int j = jl; j < LSEQ; j += JLANES) {
        const float dx = xi - sPos[j * 3 + 0];
        const float dy = yi - sPos[j * 3 + 1];
        const float dz = zi - sPos[j * 3 + 2];
        const float d  = __builtin_sqrtf(dx * dx + dy * dy + dz * dz);
#pragma unroll
        for (int r = 0; r < NUM_RBF; ++r) {
            const float t = d - c[r];
            // exp(-0.5*t*t) == 2^(t*t * NEG_HALF_LOG2E): 2 muls + v_exp_f32
            acc[r] += __builtin_amdgcn_exp2f(t * t * NEG_HALF_LOG2E);
        }
    }

#pragma unroll
    for (int r = 0; r < NUM_RBF; ++r) sPart[tid][r] = acc[r];
    __syncthreads();

    // reduce across the 16 j-lanes: thread t owns (orow, r)
    const int orow = tid >> 4;
    const int r    = tid & 15;
    float s = 0.0f;
#pragma unroll
    for (int k = 0; k < JLANES; ++k)
        s += sPart[orow * JLANES + k][r];

    const int gRow = b * LSEQ + rt * ROWS + orow;
    rbf_mean[gRow * NUM_RBF + r] = s * (1.0f / (float)LSEQ);
}

// ---------------------------------------------------------------------------
// Phase B: out[i,n] = latent[i,n] + bias[n] + sum_k rbf_mean[i,k]*W[n,k]
// via V_WMMA_F32_16X16X4_F32 chained 4x over K=16 (exact f32).
// One wave = one 16-row tile, sweeping all 16 N-tiles of HIDDEN=256.
// ---------------------------------------------------------------------------
__global__ __launch_bounds__(256) void linear_wmma_kernel(
    const float* __restrict__ rbf_mean, // [8192][16]
    const float* __restrict__ W,        // [256][16]  (W[n][k])
    const float* __restrict__ bias,     // [256]
    const float* __restrict__ latent,   // [8192][256]
    float* __restrict__ out)            // [8192][256]
{
    const int lane     = threadIdx.x & 31;
    const int waveBlk  = threadIdx.x >> 5;
    const int wave     = blockIdx.x * 8 + waveBlk;   // 0..511 row tiles
    const int rowBase  = wave * 16;

    const int m  = lane & 15;                        // row-in-tile / N-in-tile
    const int hi = lane >> 4;                        // half-wave select

    // A fragments: A[m, kc*4 + 2*hi + {0,1}] per ISA 16x4 f32 layout
    v2f a[4];
#pragma unroll
    for (int kc = 0; kc < 4; ++kc) {
        const float* p = rbf_mean + (size_t)(rowBase + m) * NUM_RBF + kc * 4 + hi * 2;
        v2f t; t.x = p[0]; t.y = p[1];
        a[kc] = t;
    }

#pragma unroll 1
    for (int nt = 0; nt < HIDDEN / 16; ++nt) {
        const int n = nt * 16 + m;

        // B fragments: B[k,n] = W[n,k]; VGPR0 holds K=2*hi, VGPR1 K=2*hi+1
        v2f bf[4];
#pragma unroll
        for (int kc = 0; kc < 4; ++kc) {
            const float* p = W + (size_t)n * NUM_RBF + kc * 4 + hi * 2;
            v2f t; t.x = p[0]; t.y = p[1];
            bf[kc] = t;
        }

        v8f cfr = {0.f, 0.f, 0.f, 0.f, 0.f, 0.f, 0.f, 0.f};
#pragma unroll
        for (int kc = 0; kc < 4; ++kc) {
            cfr = __builtin_amdgcn_wmma_f32_16x16x4_f32(
                /*neg_a=*/false, a[kc], /*neg_b=*/false, bf[kc],
                /*c_mod=*/(short)0, cfr, /*reuse_a=*/false, /*reuse_b=*/false);
        }

        const float bv = bias[n];
#pragma unroll
        for (int v = 0; v < 8; ++v) {
            const int    rrow = rowBase + v + hi * 8;      // D layout: M=v / v+8
            const int    col  = nt * 16 + m;
            const size_t idx  = (size_t)rrow * HIDDEN + col;
            out[idx] = latent[idx] + cfr[v] + bv;
        }
    }
}

// ---------------------------------------------------------------------------
extern "C" void kernel_launch(void* const* d_in, const int* in_sizes, int n_in,
                              void* d_out, int out_size, void* d_ws, size_t ws_size,
                              hipStream_t stream) {
    const float* pos     = (const float*)d_in[0];  // [8,1024,3]
    const float* latent  = (const float*)d_in[1];  // [8,1024,256]
    const float* W       = (const float*)d_in[2];  // [256,16]
    const float* bias    = (const float*)d_in[3];  // [256]
    const float* centers = (const float*)d_in[4];  // [16]
    float* out      = (float*)d_out;               // [8,1024,256]
    float* rbf_mean = (float*)d_ws;                // [8192][16] f32 = 512 KB

    rbf_mean_kernel<<<BATCH * (LSEQ / ROWS), 256, 0, stream>>>(pos, centers, rbf_mean);

    // 8192 rows / 16 per wave = 512 waves; 8 waves per 256-thread block
    linear_wmma_kernel<<<(BATCH * LSEQ / 16) / 8, 256, 0, stream>>>(
        rbf_mean, W, bias, latent, out);
}